// GCNSelf_43920335569005
// MI455X (gfx1250) — compile-verified
//
#include <hip/hip_runtime.h>

typedef __attribute__((ext_vector_type(2))) float v2f;
typedef __attribute__((ext_vector_type(8))) float v8f;

#define B_N 65536
#define NBLK 512          // B_N / (8 waves * 16 rows)
#define SX_STRIDE 66      // padded row stride for LDS staging tile

__device__ __forceinline__ void lds_fence() {
    // per-wave DS ops are in-order; this is a compiler barrier + hard wait
    asm volatile("s_wait_dscnt 0x0" ::: "memory");
}

// D = A(16x4 f32) * B(4x16 f32) + C, accumulate K=64 in 16 steps, 4 N-tiles.
__device__ __forceinline__ void wmma_gemm(const v2f a[16], const float* __restrict__ wf,
                                          int lane, v8f c[4]) {
#pragma unroll
    for (int nt = 0; nt < 4; ++nt) {
        v8f acc = {0.f, 0.f, 0.f, 0.f, 0.f, 0.f, 0.f, 0.f};
#pragma unroll
        for (int k = 0; k < 16; ++k) {
            v2f b = *(const v2f*)&wf[((k * 4 + nt) * 32 + lane) * 2];
            acc = __builtin_amdgcn_wmma_f32_16x16x4_f32(
                false, a[k], false, b, (short)0, acc, false, false);
        }
        c[nt] = acc;
    }
}

__device__ __forceinline__ void bias_relu_store(const v8f c[4], const float* __restrict__ bsh,
                                                float* __restrict__ sx, int lane) {
    int rb = (lane < 16) ? 0 : 8;
    int nc = lane & 15;
#pragma unroll
    for (int nt = 0; nt < 4; ++nt) {
        float bias = bsh[nt * 16 + nc];
#pragma unroll
        for (int j = 0; j < 8; ++j) {
            float v = c[nt][j] + bias;
            v = fmaxf(v, 0.0f);
            sx[(rb + j) * SX_STRIDE + nt * 16 + nc] = v;
        }
    }
}

__device__ __forceinline__ void load_a_from_sx(v2f a[16], const float* __restrict__ sx,
                                               int r, int off) {
#pragma unroll
    for (int k = 0; k < 16; ++k)
        a[k] = *(const v2f*)&sx[r * SX_STRIDE + 4 * k + off];
}

__global__ __launch_bounds__(256)
void gcn_fused_kernel(const int* __restrict__ user, const int* __restrict__ pos_item,
                      const int* __restrict__ neg_item,
                      const float* __restrict__ user_emb_w, const float* __restrict__ item_emb_w,
                      const float* __restrict__ W1, const float* __restrict__ b1,
                      const float* __restrict__ W2, const float* __restrict__ b2,
                      float* __restrict__ out) {
    // W fragments pre-swizzled into WMMA B-operand order:
    //   frag[((k*4 + nt)*32 + lane)*2 + t] = W[(4k + (lane<16?0:2) + t)*64 + nt*16 + lane%16]
    __shared__ float sW1f[4096];
    __shared__ float sW2f[4096];
    __shared__ float sb1[64];
    __shared__ float sb2[64];
    __shared__ float sX[8][16 * SX_STRIDE];

    const int tid = threadIdx.x;
    for (int d = tid; d < 4096; d += 256) {
        int t    = d & 1;
        int lane = (d >> 1) & 31;
        int knt  = d >> 6;
        int k    = knt >> 2;
        int nt   = knt & 3;
        int row  = 4 * k + ((lane < 16) ? 0 : 2) + t;
        int col  = nt * 16 + (lane & 15);
        sW1f[d] = W1[row * 64 + col];
        sW2f[d] = W2[row * 64 + col];
    }
    if (tid < 64) { sb1[tid] = b1[tid]; sb2[tid] = b2[tid]; }
    __syncthreads();

    const int lane = tid & 31;
    const int w    = tid >> 5;
    const int rowBase = (blockIdx.x * 8 + w) * 16;
    const int r   = lane & 15;            // tile row this lane feeds as A-operand
    const int off = (lane < 16) ? 0 : 2;  // K sub-offset per A-fragment layout
    const int nc  = lane & 15;            // N column within a 16-wide tile (C layout)
    const int rb  = (lane < 16) ? 0 : 8;  // C-layout row base for this lane half
    float* sx = &sX[w][0];

    v2f a[16];
    v8f c[4];

    // ---------------- chain A: u (== pos): x = 0.5*(user_e + pos_e) ----------------
    {
        const float* up = user_emb_w + (size_t)user[rowBase + r] * 64;
        const float* pp = item_emb_w + (size_t)pos_item[rowBase + r] * 64;
#pragma unroll
        for (int k = 0; k < 16; ++k) {
            v2f uu = *(const v2f*)(up + 4 * k + off);
            v2f vv = *(const v2f*)(pp + 4 * k + off);
            a[k] = 0.5f * (uu + vv);
        }
    }
    wmma_gemm(a, sW1f, lane, c);          // layer 1
    bias_relu_store(c, sb1, sx, lane);    // + b1, relu -> LDS staging
    lds_fence();
    load_a_from_sx(a, sx, r, off);        // re-read as layer-2 A fragments
    wmma_gemm(a, sW2f, lane, c);          // layer 2

    v8f uo[4];
#pragma unroll
    for (int nt = 0; nt < 4; ++nt) {
        float bias = sb2[nt * 16 + nc];
#pragma unroll
        for (int j = 0; j < 8; ++j) uo[nt][j] = c[nt][j] + bias;
    }

    // ---------------- chain B: neg ----------------
    {
        const float* np = item_emb_w + (size_t)neg_item[rowBase + r] * 64;
#pragma unroll
        for (int k = 0; k < 16; ++k)
            a[k] = *(const v2f*)(np + 4 * k + off);
    }
    wmma_gemm(a, sW1f, lane, c);
    bias_relu_store(c, sb1, sx, lane);
    lds_fence();
    load_a_from_sx(a, sx, r, off);
    wmma_gemm(a, sW2f, lane, c);

    v8f no[4];
#pragma unroll
    for (int nt = 0; nt < 4; ++nt) {
        float bias = sb2[nt * 16 + nc];
#pragma unroll
        for (int j = 0; j < 8; ++j) no[nt][j] = c[nt][j] + bias;
    }

    // ---------------- scores: pos_score = ||u||^2, neg_score = u . neg ----------------
    float pu[8], pn[8];
#pragma unroll
    for (int j = 0; j < 8; ++j) { pu[j] = 0.f; pn[j] = 0.f; }
#pragma unroll
    for (int nt = 0; nt < 4; ++nt)
#pragma unroll
        for (int j = 0; j < 8; ++j) {
            float uv = uo[nt][j];
            pu[j] += uv * uv;
            pn[j] += uv * no[nt][j];
        }
#pragma unroll
    for (int m = 1; m <= 8; m <<= 1) {
#pragma unroll
        for (int j = 0; j < 8; ++j) {
            pu[j] += __shfl_xor(pu[j], m, 32);
            pn[j] += __shfl_xor(pn[j], m, 32);
        }
    }

    // ---------------- stores ----------------
    float* pos_score = out;
    float* neg_score = out + B_N;
    float* concat    = out + 2 * (size_t)B_N;
    float* u_out     = out + 2 * (size_t)B_N + 128 * (size_t)B_N;

#pragma unroll
    for (int nt = 0; nt < 4; ++nt) {
        int col = nt * 16 + nc;
#pragma unroll
        for (int j = 0; j < 8; ++j) {
            size_t grow = (size_t)(rowBase + rb + j);
            float uv = uo[nt][j];
            u_out[grow * 64 + col]        = uv;  // u
            concat[grow * 128 + col]      = uv;  // pos half of concat (pos == u)
            concat[grow * 128 + 64 + col] = no[nt][j];  // neg half
        }
    }
    if (nc == 0) {
#pragma unroll
        for (int j = 0; j < 8; ++j) {
            pos_score[rowBase + rb + j] = pu[j];
            neg_score[rowBase + rb + j] = pn[j];
        }
    }
}

extern "C" void kernel_launch(void* const* d_in, const int* in_sizes, int n_in,
                              void* d_out, int out_size, void* d_ws, size_t ws_size,
                              hipStream_t stream) {
    const int*   user       = (const int*)d_in[0];
    const int*   pos_item   = (const int*)d_in[1];
    const int*   neg_item   = (const int*)d_in[2];
    const float* user_emb_w = (const float*)d_in[3];
    const float* item_emb_w = (const float*)d_in[4];
    const float* W1         = (const float*)d_in[5];
    const float* b1         = (const float*)d_in[6];
    const float* W2         = (const float*)d_in[7];
    const float* b2         = (const float*)d_in[8];
    float* out = (float*)d_out;

    gcn_fused_kernel<<<NBLK, 256, 0, stream>>>(user, pos_item, neg_item,
                                               user_emb_w, item_emb_w,
                                               W1, b1, W2, b2, out);
}